// Find_Ring_Atoms_64682207477990
// MI455X (gfx1250) — compile-verified
//
#include <hip/hip_runtime.h>
#include <stdint.h>

#define BATCH   16384
#define NATOM   50          // A
#define NNODE   51          // A+1
#define NDEG    4
#define MAXR    16

#define K1_THREADS 64
#define K2_THREADS 256
#define K2_MOLS    (K2_THREADS / MAXR)   // 16 molecules per block

// ---------------------------------------------------------------------------
// Phase 1: spanning tree + ring-edge detection. One molecule per lane.
// Edge rows async-staged into LDS; Tree/ring_idx built in LDS (byte-sized)
// and dumped with fully coalesced block-wide stores.
// ---------------------------------------------------------------------------
__global__ __launch_bounds__(K1_THREADS) void ring_phase1(
    const int* __restrict__ edge,          // (B, 50, 4) int32
    int*       __restrict__ tree_out,      // (B, 51)
    int*       __restrict__ ring_idx_out,  // (B, 51, 2)
    int*       __restrict__ nring_ws)      // (B,)
{
    __shared__ int4          me_rows[K1_THREADS * NATOM];      // 51200 B
    __shared__ unsigned char tree_s [K1_THREADS * NNODE];      //  3264 B
    __shared__ unsigned char ridx_s [K1_THREADS * NNODE * 2];  //  6528 B

    const int tid    = threadIdx.x;
    const int base_b = blockIdx.x * K1_THREADS;

    // Coalesced async stage of this block's contiguous edge tile into LDS.
    const int4* src = (const int4*)edge + (size_t)base_b * NATOM;
    for (int idx = tid; idx < K1_THREADS * NATOM; idx += K1_THREADS) {
        unsigned lds_off = (unsigned)(uintptr_t)&me_rows[idx];  // low 32b = LDS offset
        uint64_t gaddr   = (uint64_t)(uintptr_t)(src + idx);
        asm volatile("global_load_async_to_lds_b128 %0, %1, off"
                     :: "v"(lds_off), "v"(gaddr) : "memory");
    }
    // Zero the LDS output staging (dword granular; sizes are multiples of 4).
    {
        int* t32 = (int*)tree_s;
        int* r32 = (int*)ridx_s;
        for (int idx = tid; idx < (K1_THREADS * NNODE) / 4; idx += K1_THREADS)     t32[idx] = 0;
        for (int idx = tid; idx < (K1_THREADS * NNODE * 2) / 4; idx += K1_THREADS) r32[idx] = 0;
    }
    asm volatile("s_wait_asynccnt 0" ::: "memory");
    __syncthreads();

    const int b = base_b + tid;
    unsigned char* treem = tree_s + tid * NNODE;
    unsigned char* ridxm = ridx_s + tid * (NNODE * 2);

    const uint64_t FULL = (1ull << NNODE) - 1;
    uint64_t lnf = FULL & ~3ull;   // ones except nodes 0,1
    uint64_t nnv = FULL & ~1ull;   // ones except node 0
    int num_rings = 0;
    int visiting  = 1;

    for (int iter = 1; iter <= NATOM; ++iter) {
        if (iter > 1) {
            uint64_t cand = nnv & ~lnf;                 // argmax of a 0/1 vector
            visiting = cand ? (int)__builtin_ctzll(cand) : 0;
        }
        int4 row;
        if (visiting == 0) row = make_int4(-1, -1, -1, -1);   // padded row -> nxt = 0
        else               row = me_rows[tid * NATOM + (visiting - 1)];
        int comp[NDEG] = { row.x, row.y, row.z, row.w };

        #pragma unroll
        for (int d = 0; d < NDEG; ++d) {
            int nxt = comp[d] + 1;                      // me = edge + 1
            uint64_t bit = 1ull << nxt;
            bool slnf = (lnf & bit) != 0;
            bool slnv = (nnv & bit) != 0;
            bool cyc  = (!slnf) && slnv && (nxt != 0);
            if (cyc) {
                if (num_rings < NNODE) {                // JAX scatter-drop OOB
                    ridxm[num_rings * 2 + 0] = (unsigned char)visiting;
                    ridxm[num_rings * 2 + 1] = (unsigned char)nxt;
                }
                num_rings++;
            }
            if (slnf) treem[nxt] = (unsigned char)visiting;   // slot written once
            lnf &= ~bit;
        }
        nnv &= ~(1ull << visiting);
    }
    nring_ws[b] = num_rings;   // consecutive lanes -> coalesced
    __syncthreads();

    // Coalesced dump of the block's contiguous output tiles.
    int* treeg = tree_out     + (size_t)base_b * NNODE;        // 64*51 ints
    int* ridxg = ring_idx_out + (size_t)base_b * (NNODE * 2);  // 64*102 ints
    for (int idx = tid; idx < K1_THREADS * NNODE; idx += K1_THREADS)
        treeg[idx] = (int)tree_s[idx];
    for (int idx = tid; idx < K1_THREADS * NNODE * 2; idx += K1_THREADS)
        ridxg[idx] = (int)ridx_s[idx];
}

// ---------------------------------------------------------------------------
// Phase 2: per-(molecule, ring) trace through Tree parent pointers.
// Block's Tree slab async-staged into LDS; ring rows built in LDS as int8
// and dumped with coalesced block-wide stores (rings output dominates: 52 MB).
// ---------------------------------------------------------------------------
__global__ __launch_bounds__(K2_THREADS) void ring_phase2(
    const int* __restrict__ tree_in,
    const int* __restrict__ ring_idx_in,
    const int* __restrict__ nring_ws,
    int*       __restrict__ rings_out)     // (B, 16, 50), stores rings-1
{
    __shared__ int         tree32[K2_MOLS * NNODE];        // 16*51*4 = 3264 B
    __shared__ signed char ring_s[K2_THREADS * NATOM];     // 256*50  = 12800 B

    const int tid    = threadIdx.x;
    const int base_b = blockIdx.x * K2_MOLS;

    // Async stage of the block's contiguous Tree slab (16*51 ints = 204 int4).
    const int4* tsrc = (const int4*)(tree_in + (size_t)base_b * NNODE);
    for (int idx = tid; idx < (K2_MOLS * NNODE) / 4; idx += K2_THREADS) {
        unsigned lds_off = (unsigned)(uintptr_t)(((int4*)tree32) + idx);
        uint64_t gaddr   = (uint64_t)(uintptr_t)(tsrc + idx);
        asm volatile("global_load_async_to_lds_b128 %0, %1, off"
                     :: "v"(lds_off), "v"(gaddr) : "memory");
    }
    // Fill ring staging with -1 (dword granular).
    {
        int* r32 = (int*)ring_s;
        for (int idx = tid; idx < (K2_THREADS * NATOM) / 4; idx += K2_THREADS)
            r32[idx] = -1;
    }
    asm volatile("s_wait_asynccnt 0" ::: "memory");
    __syncthreads();

    const int mb = tid >> 4;          // molecule within block (0..15)
    const int i  = tid & (MAXR - 1);  // ring slot 0..15
    const int b  = base_b + mb;
    signed char* row = ring_s + tid * NATOM;   // this thread's 50-entry row

    int nr = nring_ws[b];
    if (i < nr) {                                   // i >= 16 writes are dropped anyway
        const int* tr = tree32 + mb * NNODE;
        int r0 = ring_idx_in[(size_t)b * (2 * NNODE) + i * 2 + 0];
        int r1 = ring_idx_in[(size_t)b * (2 * NNODE) + i * 2 + 1];

        // mark0: mark full root-path from r0
        int s = r0;
        uint64_t vb0 = 1ull << s;
        for (int k = 0; k < NATOM; ++k) { s = tr[s]; vb0 |= 1ull << s; }

        // b1: walk from r1, emit until path meets vb0
        s = r1;
        uint64_t vb1 = 1ull << s;
        int air = 0;
        row[0] = (signed char)(s - 1);              // rings.set(start), output is rings-1
        for (int k = 0; k < NATOM; ++k) {
            int nxt = tr[s];
            bool same = (vb0 & vb1) != 0;
            if (!same) { air++; if (air < NATOM) row[air] = (signed char)(nxt - 1); }
            vb1 |= 1ull << nxt;
            s = nxt;
        }

        // bridge: restart from r0
        s = r0;
        uint64_t vb0b = 1ull << s;
        air++;
        if (air < NATOM) row[air] = (signed char)(s - 1);

        // b2: walk from r0, emit until it meets vb1
        for (int k = 0; k < NATOM; ++k) {
            int nxt = tr[s];
            vb0b |= 1ull << nxt;
            bool same = (vb0b & vb1) != 0;
            if (!same) { air++; if (air < NATOM) row[air] = (signed char)(nxt - 1); }
            s = nxt;
        }
    }
    __syncthreads();

    // Coalesced dump: block's rings region is contiguous (256 rows * 50 ints).
    int* rblock = rings_out + (size_t)blockIdx.x * (K2_THREADS * NATOM);
    for (int idx = tid; idx < K2_THREADS * NATOM; idx += K2_THREADS)
        rblock[idx] = (int)ring_s[idx];
}

// ---------------------------------------------------------------------------
extern "C" void kernel_launch(void* const* d_in, const int* in_sizes, int n_in,
                              void* d_out, int out_size, void* d_ws, size_t ws_size,
                              hipStream_t stream) {
    (void)in_sizes; (void)n_in; (void)out_size; (void)ws_size;
    const int* edge = (const int*)d_in[0];

    int* out          = (int*)d_out;
    int* tree_out     = out;                                      // B*51
    int* ring_idx_out = out + (size_t)BATCH * NNODE;              // B*51*2
    int* rings_out    = ring_idx_out + (size_t)BATCH * NNODE * 2; // B*16*50
    int* nring_ws     = (int*)d_ws;                               // B ints

    ring_phase1<<<BATCH / K1_THREADS, K1_THREADS, 0, stream>>>(
        edge, tree_out, ring_idx_out, nring_ws);
    ring_phase2<<<BATCH / K2_MOLS, K2_THREADS, 0, stream>>>(
        tree_out, ring_idx_out, nring_ws, rings_out);
}